// SoftDynamicTimeWarping_29343216566480
// MI455X (gfx1250) — compile-verified
//
#include <hip/hip_runtime.h>

#ifndef __has_builtin
#define __has_builtin(x) 0
#endif

namespace {

constexpr int kN = 512;          // rows (ii = 1..512)
constexpr int kM = 512;          // cols (jj = 1..512)
constexpr int kTPB = 512;        // one thread per row
constexpr int kRowStride = 10;   // 8 data floats + 2 pad (8B-aligned, bank-skewed)
constexpr float kGamma = 0.1f;
constexpr float kInvGamma = 10.0f;

#define SDTW_INF __builtin_huge_valf()

// v2i matches the builtin's "int __vector(2)" parameter pointee type.
typedef int v2i __attribute__((vector_size(2 * sizeof(int))));
typedef __attribute__((address_space(1))) v2i gv2i;  // global (prints as __device__)
typedef __attribute__((address_space(3))) v2i lv2i;  // LDS

// ---- CDNA5 async global->LDS copy (8 bytes), ASYNCcnt-tracked ----
__device__ __forceinline__ void asyncCopyB64(const float* gsrc, float* ldst) {
#if __has_builtin(__builtin_amdgcn_global_load_async_to_lds_b64)
  __builtin_amdgcn_global_load_async_to_lds_b64((gv2i*)gsrc, (lv2i*)ldst,
                                                0 /*imm offset*/, 0 /*cpol*/);
#else
  // synchronous fallback (correct, just not async)
  ((float2*)ldst)[0] = ((const float2*)gsrc)[0];
#endif
}

__device__ __forceinline__ void waitAsyncLe0() {
#if __has_builtin(__builtin_amdgcn_s_wait_asynccnt)
  __builtin_amdgcn_s_wait_asynccnt(0);
#else
  asm volatile("s_wait_asynccnt 0" ::: "memory");
#endif
  asm volatile("" ::: "memory");  // keep later LDS reads below the wait
}

__device__ __forceinline__ void waitAsyncLe4() {
#if __has_builtin(__builtin_amdgcn_s_wait_asynccnt)
  __builtin_amdgcn_s_wait_asynccnt(4);
#else
  asm volatile("s_wait_asynccnt 4" ::: "memory");
#endif
  asm volatile("" ::: "memory");
}

__global__ __launch_bounds__(kTPB, 1) void softdtw_kernel(
    const float* __restrict__ D, const float* __restrict__ mask,
    const int* __restrict__ xLens, const int* __restrict__ yLens,
    float* __restrict__ out) {
  // Two ping-pong diagonals indexed by ii = 0..512 (entry 0 stays +inf forever).
  __shared__ float diag[2][kTPB + 2];
  // Skewed per-thread D staging: 2 chunk buffers x 512 rows x (8+2) floats.
  __shared__ float dtile[2][kTPB * kRowStride];
  __shared__ float scratch[8];  // sink for cadence-keeping dummy async loads

  const int t = threadIdx.x;          // row index (ii = t+1)
  const int b = blockIdx.x;           // batch
  const int xl = xLens[b];
  const int yl = yLens[b];
  const int sOut = xl + yl;           // diagonal holding the loss cell

  const float* Drow = D + ((size_t)b * kN + t) * kM;
  const float* Mrow = mask + ((size_t)b * kN + t) * kM;

  // ---- init: d_{s-1} = all +inf; boundary entries [0] stay +inf ----
  diag[0][t + 1] = SDTW_INF;
  if (t == 0) { diag[0][0] = SDTW_INF; diag[1][0] = SDTW_INF; }

  // ---- preload D chunks 0 and 1 (cols 0..15) via async-to-LDS ----
  {
    float* l0 = &dtile[0][t * kRowStride];
    float* l1 = &dtile[1][t * kRowStride];
    asyncCopyB64(Drow + 0, l0 + 0);
    asyncCopyB64(Drow + 2, l0 + 2);
    asyncCopyB64(Drow + 4, l0 + 4);
    asyncCopyB64(Drow + 6, l0 + 6);
    asyncCopyB64(Drow + 8,  l1 + 0);
    asyncCopyB64(Drow + 10, l1 + 2);
    asyncCopyB64(Drow + 12, l1 + 4);
    asyncCopyB64(Drow + 14, l1 + 6);
  }
  waitAsyncLe0();
  __syncthreads();

  float my_dm1 = SDTW_INF;                       // d_{s-1}[ii]   (register)
  float nb_dm2 = (t == 0) ? 0.0f : SDTW_INF;     // d_{s-2}[ii-1] (register; d0[0]=0)
  float4 mbuf = make_float4(1.f, 1.f, 1.f, 1.f); // mask register stream
  int p = 0;

  for (int s = 2; s <= kN + kM; ++s) {
    const int c = s - 2 - t;                     // column = jj-1
    const bool inrange = (c >= 0) && (c < kM);

    // Chunk boundary: ensure this chunk's async loads have landed.
    // (exactly 4 async instrs (next chunk) are issued after a chunk before its use,
    //  and async loads complete in order, so "<= 4 outstanding" suffices.)
    if (inrange && c >= 16 && (c & 7) == 0) waitAsyncLe4();

    float dval = 0.0f, mval = 0.0f;
    if (inrange) {
      if ((c & 3) == 0) {
        mbuf = *(const float4*)(Mrow + c);
        if ((c & 31) == 0 && c + 64 < kM)
          __builtin_prefetch(Mrow + c + 64, 0, 0);   // global_prefetch_b8
      }
      dval = dtile[(c >> 3) & 1][t * kRowStride + (c & 7)];
      const int q = c & 3;
      mval = (q == 0) ? mbuf.x : (q == 1) ? mbuf.y : (q == 2) ? mbuf.z : mbuf.w;
    }

    const float nb_dm1 = diag[p][t];             // d_{s-1}[ii-1] via LDS

    // softmin with gamma = 0.1 (PENALTY = 0), max-shifted LSE, inf-safe.
    const float r0 = -nb_dm2 * kInvGamma;
    const float r1 = -nb_dm1 * kInvGamma;
    const float r2 = -my_dm1 * kInvGamma;
    const float rmax = fmaxf(r0, fmaxf(r1, r2));
    const bool fin = (rmax > -SDTW_INF);
    const float rs = fin ? rmax : 0.0f;
    const float rsum = __expf(r0 - rs) + __expf(r1 - rs) + __expf(r2 - rs);
    const float softmin = fin ? (-kGamma * (__logf(rsum) + rs)) : SDTW_INF;

    const bool upd = inrange && (t < xl) && (c < yl);  // ii<=xl, jj<=yl
    const float dnew = upd ? (dval * mval + softmin) : SDTW_INF;

    if (s == sOut && t == xl - 1) out[b] = dnew;       // loss cell (xl, yl)

    diag[p ^ 1][t + 1] = dnew;

    // Issue async prefetch for chunk k+2; keep a uniform issue cadence past
    // the stream tail with clamped dummy loads into scratch.
    if ((c & 7) == 7 && c >= 7 && c < 552) {
      asm volatile("" ::: "memory");  // keep issues after this step's LDS read
      const int k2 = (c >> 3) + 2;
      if (k2 < (kM >> 3)) {
        float* l = &dtile[k2 & 1][t * kRowStride];
        const float* g = Drow + (k2 << 3);
        asyncCopyB64(g + 0, l + 0);
        asyncCopyB64(g + 2, l + 2);
        asyncCopyB64(g + 4, l + 4);
        asyncCopyB64(g + 6, l + 6);
      } else {
        const float* g = Drow + (kM - 8);
        asyncCopyB64(g + 0, scratch + 0);
        asyncCopyB64(g + 2, scratch + 2);
        asyncCopyB64(g + 4, scratch + 4);
        asyncCopyB64(g + 6, scratch + 6);
      }
    }

    __syncthreads();      // split barrier: publishes diag[p^1], protects ping-pong
    p ^= 1;
    my_dm1 = dnew;        // becomes d_{s-1}[ii]
    nb_dm2 = nb_dm1;      // becomes d_{s-2}[ii-1]
  }
}

}  // namespace

extern "C" void kernel_launch(void* const* d_in, const int* in_sizes, int n_in,
                              void* d_out, int out_size, void* d_ws, size_t ws_size,
                              hipStream_t stream) {
  (void)in_sizes; (void)n_in; (void)d_ws; (void)ws_size; (void)out_size;
  const float* D    = (const float*)d_in[0];
  const float* mask = (const float*)d_in[1];
  const int* xLens  = (const int*)d_in[2];
  const int* yLens  = (const int*)d_in[3];
  float* out        = (float*)d_out;
  softdtw_kernel<<<dim3(64), dim3(kTPB), 0, stream>>>(D, mask, xLens, yLens, out);
}